// DiceAndCrossEntropyLoss_11587821765242
// MI455X (gfx1250) — compile-verified
//
#include <hip/hip_runtime.h>

// Problem constants
#define B_   16
#define C_   8
#define H_   512
#define W_   512
#define HW_  (H_*W_)          // 262144 pixels per (b,c) plane
#define BLK_PER_B 64
#define THREADS   256
#define NPART (BLK_PER_B*B_)  // 1024 block-partials per quantity
#define NQ    25              // tp[0..7], psum[0..7], cnt[0..7], ce
#define SMOOTH 1e-5f
#define EPSV   1e-8f

typedef float v2f __attribute__((ext_vector_type(2)));
typedef float v8f __attribute__((ext_vector_type(8)));

#define COMP(v4, j) ((j)==0 ? (v4).x : (j)==1 ? (v4).y : (j)==2 ? (v4).z : (v4).w)

__device__ __forceinline__ float wred32(float v) {
#pragma unroll
  for (int off = 16; off > 0; off >>= 1) v += __shfl_xor(v, off, 32);
  return v;
}

// ---------------------------------------------------------------------------
// Kernel 1: stream x/y once, accumulate 25 quantities per thread, reduce to
// one partial per (quantity, block). Deterministic (no float atomics).
// ---------------------------------------------------------------------------
__global__ void dice_ce_partial(const float* __restrict__ x,
                                const int*   __restrict__ y,
                                float*       __restrict__ partials) {
  const int b   = blockIdx.y;
  const int tid = blockIdx.x * blockDim.x + threadIdx.x;      // 0..16383
  const float* xb = x + (size_t)b * (size_t)(C_ * HW_);
  const int*   yb = y + (size_t)b * (size_t)HW_;

  float acc[NQ];
#pragma unroll
  for (int i = 0; i < NQ; i++) acc[i] = 0.f;

  // 4 pixels per iteration: 8x float4 (one per channel) + 1x int4 labels.
  for (int g = tid; g < HW_ / 4; g += BLK_PER_B * THREADS) {
    const int pix = g * 4;
    float4 xc[C_];
#pragma unroll
    for (int c = 0; c < C_; c++)
      xc[c] = *reinterpret_cast<const float4*>(xb + (size_t)c * HW_ + pix);
    int4 y4 = *reinterpret_cast<const int4*>(yb + pix);

#pragma unroll
    for (int j = 0; j < 4; j++) {
      float v[C_];
#pragma unroll
      for (int c = 0; c < C_; c++) v[c] = COMP(xc[c], j);
      const int lab = COMP(y4, j);

      float m = v[0];
#pragma unroll
      for (int c = 1; c < C_; c++) m = fmaxf(m, v[c]);

      float e[C_];
      float s = 0.f;
#pragma unroll
      for (int c = 0; c < C_; c++) { e[c] = __expf(v[c] - m); s += e[c]; }
      const float inv = 1.0f / s;
      const float lse = __logf(s);

      float xl = 0.f;
#pragma unroll
      for (int c = 0; c < C_; c++) {
        const float p   = e[c] * inv;
        const bool  hit = (lab == c);
        acc[8 + c]  += p;                    // p_sum[c]
        acc[c]      += hit ? p   : 0.f;      // tp[c]
        acc[16 + c] += hit ? 1.f : 0.f;      // counts[c]
        xl          += hit ? v[c] : 0.f;     // x[label]
      }
      acc[24] += m + lse - xl;               // nll
    }
  }

  // wave32 shuffle reduction -> LDS -> per-block partials
  __shared__ float lds[THREADS / 32][NQ];
  const int wave = threadIdx.x >> 5;
  const int lane = threadIdx.x & 31;
#pragma unroll
  for (int i = 0; i < NQ; i++) {
    const float r = wred32(acc[i]);
    if (lane == 0) lds[wave][i] = r;
  }
  __syncthreads();
  if (threadIdx.x < NQ) {
    float sum = 0.f;
#pragma unroll
    for (int w = 0; w < THREADS / 32; w++) sum += lds[w][threadIdx.x];
    partials[threadIdx.x * NPART + b * BLK_PER_B + blockIdx.x] = sum;
  }
}

// ---------------------------------------------------------------------------
// Kernel 2: reduce block partials; compute 128 dice values and sum them with
// V_WMMA_F32_16X16X4_F32 (A = dice values in documented 16x4 f32 A-layout,
// B = all-ones so the layout is irrelevant, two WMMAs chained through C).
// ---------------------------------------------------------------------------
__global__ void dice_ce_final(const float* __restrict__ partials,
                              float*       __restrict__ out) {
  __shared__ float red[24][16];   // red[q][b]: q 0..7 tp, 8..15 psum, 16..23 cnt
  __shared__ float cep[128];
  const int t = threadIdx.x;

  if (t < 384) {
    const int q = t >> 4, b = t & 15;
    const float* p = partials + q * NPART + b * BLK_PER_B;
    float s = 0.f;
    for (int i = 0; i < BLK_PER_B; i++) s += p[i];
    red[q][b] = s;
  } else {
    const int j = t - 384;                      // 0..127
    const float* p = partials + 24 * NPART + j * 8;
    float s = 0.f;
    for (int i = 0; i < 8; i++) s += p[i];
    cep[j] = s;
  }
  __syncthreads();

  if (t < 32) {                                 // wave 0, EXEC all ones
    const int m  = t & 15;
    const int kb = (t >> 4) * 2;                // K pair handled by this lane

    auto dcval = [&](int idx) -> float {        // idx = b*8 + c, 0..127
      const int b = idx >> 3, c = idx & 7;
      if (c == 0) return 0.f;                   // class 0 excluded from mean
      const float tp = red[c][b];
      const float ps = red[8 + c][b];
      const float cn = red[16 + c][b];
      return (2.f * tp + SMOOTH) / (ps + cn + SMOOTH + EPSV);
    };

    // A-matrix 16x4 f32 layout: element (M,K) -> lane = M + 16*(K>=2), vgpr = K&1
    v2f a0, a1, ones;
    a0.x = dcval(m * 4 + kb);
    a0.y = dcval(m * 4 + kb + 1);
    a1.x = dcval(64 + m * 4 + kb);
    a1.y = dcval(64 + m * 4 + kb + 1);
    ones.x = 1.f; ones.y = 1.f;

    v8f c = {};
    c = __builtin_amdgcn_wmma_f32_16x16x4_f32(false, a0, false, ones, (short)0, c, false, false);
    c = __builtin_amdgcn_wmma_f32_16x16x4_f32(false, a1, false, ones, (short)0, c, false, false);

    // D rows are replicated across columns; lanes 0-15 hold rows 0-7 in c[0..7],
    // lanes 16-31 hold rows 8-15. Sum the 8 rows, then merge halves.
    float half  = c[0] + c[1] + c[2] + c[3] + c[4] + c[5] + c[6] + c[7];
    float dcsum = half + __shfl_xor(half, 16, 32);

    float ce4   = cep[t] + cep[t + 32] + cep[t + 64] + cep[t + 96];
    float cesum = wred32(ce4);

    if (t == 0) {
      const float dc_loss = 1.f - dcsum / 112.f;                 // 16 * 7 classes
      const float ce_loss = cesum / (float)(B_ * HW_);
      out[0] = dc_loss + ce_loss;                                // weights = 1.0
    }
  }
}

// ---------------------------------------------------------------------------
extern "C" void kernel_launch(void* const* d_in, const int* in_sizes, int n_in,
                              void* d_out, int out_size, void* d_ws, size_t ws_size,
                              hipStream_t stream) {
  (void)in_sizes; (void)n_in; (void)out_size; (void)ws_size;
  const float* x = (const float*)d_in[0];
  const int*   y = (const int*)d_in[1];
  float* partials = (float*)d_ws;          // needs 25*1024 floats = 100 KiB
  float* out = (float*)d_out;

  dice_ce_partial<<<dim3(BLK_PER_B, B_), THREADS, 0, stream>>>(x, y, partials);
  dice_ce_final<<<1, 512, 0, stream>>>(partials, out);
}